// GNNLayerAttention_84181359002189
// MI455X (gfx1250) — compile-verified
//
#include <hip/hip_runtime.h>
#include <hip/hip_bf16.h>
#include <math.h>

typedef __attribute__((ext_vector_type(2))) float v2f;
typedef __attribute__((ext_vector_type(8))) float v8f;

#define DFEAT 64
#define NEG_SLOPE 0.2f

__device__ __forceinline__ float leaky(float x) { return x >= 0.0f ? x : NEG_SLOPE * x; }

// Exact float atomic max via sign-split int/uint atomics (init must be -inf).
__device__ __forceinline__ void atomicMaxF(float* addr, float v) {
    if (v >= 0.0f) atomicMax((int*)addr, __float_as_int(v));
    else           atomicMin((unsigned int*)addr, __float_as_uint(v));
}

// ---------------------------------------------------------------------------
// Kernel 1: v1 = a1 @ W_att, v2 = a2 @ W_att, c1 = b.a1, c2 = b.a2
// ---------------------------------------------------------------------------
__global__ void gnn_prep_kernel(const float* __restrict__ Watt,
                                const float* __restrict__ batt,
                                const float* __restrict__ avec,
                                float* __restrict__ v12) {
    int k = threadIdx.x;  // 64 threads
    float acc1 = 0.0f, acc2 = 0.0f;
    for (int j = 0; j < DFEAT; ++j) {
        float w = Watt[j * DFEAT + k];
        acc1 += avec[j] * w;
        acc2 += avec[DFEAT + j] * w;
    }
    v12[k] = acc1;
    v12[DFEAT + k] = acc2;
    if (k == 0) {
        float c1 = 0.0f, c2 = 0.0f;
        for (int j = 0; j < DFEAT; ++j) {
            c1 += batt[j] * avec[j];
            c2 += batt[j] * avec[DFEAT + j];
        }
        v12[128] = c1;
        v12[129] = c2;
    }
}

// ---------------------------------------------------------------------------
// Kernel 2: zero h_neigh / denom, max_score = -inf
// ---------------------------------------------------------------------------
__global__ void gnn_init_kernel(float* __restrict__ h_nei,
                                float* __restrict__ maxsc,
                                float* __restrict__ denom, int N) {
    long i = (long)blockIdx.x * blockDim.x + threadIdx.x;
    if (i < (long)N * DFEAT) h_nei[i] = 0.0f;
    if (i < N) { maxsc[i] = -INFINITY; denom[i] = 0.0f; }
}

// ---------------------------------------------------------------------------
// Kernel 3: per-node attention pre-scores (64-dot per node, float4 loads)
// ---------------------------------------------------------------------------
__global__ void gnn_node_scores_kernel(const float* __restrict__ feat,
                                       const float* __restrict__ v12,
                                       float* __restrict__ s_src,
                                       float* __restrict__ s_dst, int N) {
    int n = blockIdx.x * blockDim.x + threadIdx.x;
    if (n >= N) return;
    const float4* f4 = (const float4*)(feat + (size_t)n * DFEAT);
    const float4* v1 = (const float4*)v12;
    const float4* v2 = (const float4*)(v12 + DFEAT);
    float a1 = 0.0f, a2 = 0.0f;
#pragma unroll
    for (int t = 0; t < DFEAT / 4; ++t) {
        float4 f = f4[t], x = v1[t], y = v2[t];
        a1 += f.x * x.x + f.y * x.y + f.z * x.z + f.w * x.w;
        a2 += f.x * y.x + f.y * y.y + f.z * y.z + f.w * y.w;
    }
    s_src[n] = a1 + v12[128];
    s_dst[n] = a2 + v12[129];
}

// ---------------------------------------------------------------------------
// Kernel 4: h_msg = feat @ W1^T + b1 via V_WMMA_F32_16X16X4_F32.
// One wave32 per 16-row tile, 4 accumulators cover all 64 output columns,
// K-loop of 16 steps of 4. A layout: lane&15 = M, lane>>4 selects K pair
// {0,1}/{2,3}. B layout mirrors (rows striped across lanes per VGPR).
// ---------------------------------------------------------------------------
__global__ void gnn_hmsg_wmma_kernel(const float* __restrict__ feat,
                                     const float* __restrict__ W1,
                                     const float* __restrict__ b1,
                                     float* __restrict__ h_msg,
                                     int N, int ntiles) {
    int wave = threadIdx.x >> 5;
    int lane = threadIdx.x & 31;
    int tile = blockIdx.x * (blockDim.x >> 5) + wave;
    if (tile >= ntiles) return;  // wave-uniform: EXEC stays all-ones for WMMA
    int row0 = tile << 4;
    int half = lane >> 4;   // 0|1
    int l15  = lane & 15;

    v8f acc[4] = {};
    int rowA = row0 + l15;
    if (rowA >= N) rowA = N - 1;  // clamp (stores guarded below)
    const float* arow = feat + (size_t)rowA * DFEAT;

#pragma unroll
    for (int k = 0; k < DFEAT; k += 4) {
        int kk = k + (half << 1);
        v2f a;
        a.x = arow[kk];
        a.y = arow[kk + 1];
#pragma unroll
        for (int c = 0; c < 4; ++c) {
            int j = (c << 4) + l15;                 // output column
            const float* wr = W1 + (size_t)j * DFEAT + kk;  // B[k][j] = W1[j][k]
            v2f b;
            b.x = wr[0];
            b.y = wr[1];
            acc[c] = __builtin_amdgcn_wmma_f32_16x16x4_f32(
                false, a, false, b, (short)0, acc[c], false, false);
        }
    }
#pragma unroll
    for (int c = 0; c < 4; ++c) {
        int col = (c << 4) + l15;
        float bias = b1[col];
#pragma unroll
        for (int g = 0; g < 8; ++g) {
            int row = row0 + g + (half << 3);
            if (row < N) h_msg[(size_t)row * DFEAT + col] = acc[c][g] + bias;
        }
    }
}

// ---------------------------------------------------------------------------
// Kernel 5: per-edge raw score + leaky relu, atomic segment max over dst
// ---------------------------------------------------------------------------
__global__ void gnn_edge_scores_kernel(const int* __restrict__ src,
                                       const int* __restrict__ dst,
                                       const int* __restrict__ etype,
                                       const float* __restrict__ s_src,
                                       const float* __restrict__ s_dst,
                                       float* __restrict__ e_buf,
                                       float* __restrict__ maxsc, int E) {
    int e = blockIdx.x * blockDim.x + threadIdx.x;
    if (e >= E) return;
    int s = src[e], d = dst[e];
    float raw = s_src[s] + s_dst[d] + (etype[e] == 0 ? 5.0f : 0.0f);
    float val = leaky(raw * 0.125f);  // scaling = 1/sqrt(64)
    e_buf[e] = val;
    atomicMaxF(&maxsc[d], val);
}

// ---------------------------------------------------------------------------
// Kernel 6: exp(e - max[dst]) in-place, atomic segment sum of denom
// ---------------------------------------------------------------------------
__global__ void gnn_edge_exp_kernel(const int* __restrict__ dst,
                                    float* __restrict__ e_buf,
                                    const float* __restrict__ maxsc,
                                    float* __restrict__ denom, int E) {
    int e = blockIdx.x * blockDim.x + threadIdx.x;
    if (e >= E) return;
    int d = dst[e];
    float ex = expf(e_buf[e] - maxsc[d]);
    e_buf[e] = ex;
    atomicAdd(&denom[d], ex);
}

// ---------------------------------------------------------------------------
// Kernel 7: h_neigh[dst] += alpha * h_msg[src]. One wave per edge, each lane
// handles 2 features (float2 gather, 2 f32 atomics). h_msg/h_neigh both
// L2-resident (12.8 MB each in 192 MB L2).
// ---------------------------------------------------------------------------
__global__ void gnn_aggregate_kernel(const int* __restrict__ src,
                                     const int* __restrict__ dst,
                                     const float* __restrict__ e_buf,
                                     const float* __restrict__ denom,
                                     const float* __restrict__ h_msg,
                                     float* __restrict__ h_nei, int E) {
    long gid = (long)blockIdx.x * blockDim.x + threadIdx.x;
    long w = gid >> 5;
    if (w >= E) return;
    int e = (int)w;
    int lane = (int)(gid & 31);
    int s = src[e], d = dst[e];
    float alpha = e_buf[e] / (denom[d] + 1e-9f);
    const float* ms = h_msg + (size_t)s * DFEAT + lane * 2;
    float* hn = h_nei + (size_t)d * DFEAT + lane * 2;
    atomicAdd(hn,     alpha * ms[0]);
    atomicAdd(hn + 1, alpha * ms[1]);
}

// ---------------------------------------------------------------------------
// Kernel 8: out = leaky(feat + h_neigh + (feat*h_neigh) @ W2^T + b2)
// WMMA GEMM with fused elementwise A construction and epilogue.
// ---------------------------------------------------------------------------
__global__ void gnn_finalize_wmma_kernel(const float* __restrict__ feat,
                                         const float* __restrict__ h_nei,
                                         const float* __restrict__ W2,
                                         const float* __restrict__ b2,
                                         float* __restrict__ out,
                                         int N, int ntiles) {
    int wave = threadIdx.x >> 5;
    int lane = threadIdx.x & 31;
    int tile = blockIdx.x * (blockDim.x >> 5) + wave;
    if (tile >= ntiles) return;
    int row0 = tile << 4;
    int half = lane >> 4;
    int l15  = lane & 15;

    v8f acc[4] = {};
    int rowA = row0 + l15;
    if (rowA >= N) rowA = N - 1;
    const float* frow = feat  + (size_t)rowA * DFEAT;
    const float* nrow = h_nei + (size_t)rowA * DFEAT;

#pragma unroll
    for (int k = 0; k < DFEAT; k += 4) {
        int kk = k + (half << 1);
        v2f a;
        a.x = frow[kk]     * nrow[kk];       // prod_h on the fly
        a.y = frow[kk + 1] * nrow[kk + 1];
#pragma unroll
        for (int c = 0; c < 4; ++c) {
            int j = (c << 4) + l15;
            const float* wr = W2 + (size_t)j * DFEAT + kk;
            v2f b;
            b.x = wr[0];
            b.y = wr[1];
            acc[c] = __builtin_amdgcn_wmma_f32_16x16x4_f32(
                false, a, false, b, (short)0, acc[c], false, false);
        }
    }
#pragma unroll
    for (int c = 0; c < 4; ++c) {
        int col = (c << 4) + l15;
        float bias = b2[col];
#pragma unroll
        for (int g = 0; g < 8; ++g) {
            int row = row0 + g + (half << 3);
            if (row < N) {
                size_t idx = (size_t)row * DFEAT + col;
                float f = feat[idx], h = h_nei[idx];
                out[idx] = leaky(f + h + acc[c][g] + bias);
            }
        }
    }
}

// ---------------------------------------------------------------------------
extern "C" void kernel_launch(void* const* d_in, const int* in_sizes, int n_in,
                              void* d_out, int out_size, void* d_ws, size_t ws_size,
                              hipStream_t stream) {
    const int*   indices = (const int*)d_in[0];   // (2, E)
    const float* feat    = (const float*)d_in[1]; // (N, 64)
    const int*   etype   = (const int*)d_in[2];   // (E,)
    // d_in[3] = num_nodes scalar (derived from sizes instead)
    const float* W1   = (const float*)d_in[4];
    const float* b1   = (const float*)d_in[5];
    const float* W2   = (const float*)d_in[6];
    const float* b2   = (const float*)d_in[7];
    const float* Watt = (const float*)d_in[8];
    const float* batt = (const float*)d_in[9];
    const float* avec = (const float*)d_in[10];   // (128,1)

    const int N = in_sizes[1] / DFEAT;
    const int E = in_sizes[2];
    const int* src = indices;
    const int* dst = indices + E;

    // Workspace layout (floats), all offsets 16B-aligned.
    float* ws = (float*)d_ws;
    size_t Np = ((size_t)N + 63) & ~(size_t)63;
    float* v12   = ws;                 // 192 floats: v1(64), v2(64), c1, c2
    float* s_src = ws + 192;           // Np
    float* s_dst = s_src + Np;         // Np
    float* maxsc = s_dst + Np;         // Np
    float* denom = maxsc + Np;         // Np
    float* h_msg = denom + Np;         // Np*64
    float* h_nei = h_msg + Np * DFEAT; // Np*64
    float* e_buf = h_nei + Np * DFEAT; // E

    gnn_prep_kernel<<<1, 64, 0, stream>>>(Watt, batt, avec, v12);

    {
        long tot = (long)N * DFEAT;
        gnn_init_kernel<<<(int)((tot + 255) / 256), 256, 0, stream>>>(h_nei, maxsc, denom, N);
    }

    gnn_node_scores_kernel<<<(N + 255) / 256, 256, 0, stream>>>(feat, v12, s_src, s_dst, N);

    const int ntiles = (N + 15) / 16;
    const int wmma_blocks = (ntiles + 7) / 8;   // 8 waves (wave32) per 256-thread block
    gnn_hmsg_wmma_kernel<<<wmma_blocks, 256, 0, stream>>>(feat, W1, b1, h_msg, N, ntiles);

    gnn_edge_scores_kernel<<<(E + 255) / 256, 256, 0, stream>>>(src, dst, etype, s_src, s_dst,
                                                                e_buf, maxsc, E);
    gnn_edge_exp_kernel<<<(E + 255) / 256, 256, 0, stream>>>(dst, e_buf, maxsc, denom, E);

    {
        long tot = (long)E * 32;  // one wave per edge
        gnn_aggregate_kernel<<<(int)((tot + 255) / 256), 256, 0, stream>>>(src, dst, e_buf, denom,
                                                                           h_msg, h_nei, E);
    }

    gnn_finalize_wmma_kernel<<<wmma_blocks, 256, 0, stream>>>(feat, h_nei, W2, b2,
                                                              (float*)d_out, N, ntiles);
}